// bicon_loss_71588514889841
// MI455X (gfx1250) — compile-verified
//
#include <hip/hip_runtime.h>
#include <hip/hip_bf16.h>

// ---------------------------------------------------------------------------
// Fused "bicon loss" for B=8, C=8, H=512, W=512 fp32 inputs.
//  * single pass over atts/dets/con_target/target (~200 MiB -> ~12 us roofline)
//  * LDS-tiled sigmoid with 1-px halo; center sigmoids kept in registers,
//    LDS read only for the 8 neighbor taps
//  * staging batches 8 channel loads per thread before waiting (MLP), with
//    the bounds check hoisted out of the load batch
//  * sigmoid reciprocal via v_rcp_f32; exp via v_exp_f32
//  * BCE terms use the exact binary-target identity
//      t*clog(p) + (1-t)*clog(1-p) == clog(t>0.5 ? p : 1-p)   for t in {0,1}
//  * wave32 reduction via V_WMMA_F32_16X16X4_F32 ones-matrix trick
//  * deterministic two-level block reduction through d_ws
// ---------------------------------------------------------------------------

#define BATCH  8
#define CHANS  8
#define HH     512
#define WW     512
#define TS     16            // tile side
#define HALO   (TS + 2)      // 18x18 halo tile
#define RING   (4 * HALO - 4)                     // 68 perimeter cells
#define NBLOCKS ((WW / TS) * (HH / TS) * BATCH)   // 32*32*8 = 8192

typedef __attribute__((ext_vector_type(2))) float v2f;
typedef __attribute__((ext_vector_type(8))) float v8f;

// BCE element term for binary targets t in {0,1} (exact: the discarded side
// of t*clog(p)+(1-t)*clog(1-p) is multiplied by 0.0 after the -100 clamp).
__device__ __forceinline__ float bce_term(float p, float t) {
    const float q = (t > 0.5f) ? p : (1.0f - p);
    return fmaxf(__logf(q), -100.0f);
}

// sigmoid with v_rcp_f32 instead of IEEE divide (1 trans op; the loss is a
// 16M-element mean, far below this ~1 ulp noise floor)
__device__ __forceinline__ float fast_sigmoid(float x) {
    return __builtin_amdgcn_rcpf(1.0f + __expf(-x));
}

// Full-wave (32 lane) sum via V_WMMA_F32_16X16X4_F32.
// A[m,k]: lane m (0..15) -> A[m,0]=x, lane m+16 -> A[m,2]=x (other K slots 0).
// B = all ones =>  D[m,n] = x_m + x_{m+16}.
// D vgpr j: lanes 0..15 hold M=j, lanes 16..31 hold M=j+8, so summing the 8
// D registers gives each half-wave's total; shfl_xor(16) completes the sum.
__device__ __forceinline__ float wave_sum_wmma(float x) {
    v2f a; a.x = x;    a.y = 0.0f;
    v2f b; b.x = 1.0f; b.y = 1.0f;
    v8f c = {};
    v8f d = __builtin_amdgcn_wmma_f32_16x16x4_f32(
        /*neg_a=*/false, a, /*neg_b=*/false, b,
        /*c_mod=*/(short)0, c, /*reuse_a=*/false, /*reuse_b=*/false);
    float s = ((d[0] + d[1]) + (d[2] + d[3])) + ((d[4] + d[5]) + (d[6] + d[7]));
    s += __shfl_xor(s, 16, 32);
    return s;
}

__global__ __launch_bounds__(256) void bicon_loss_fused_kernel(
    const float* __restrict__ atts,
    const float* __restrict__ dets,
    const float* __restrict__ target,
    const float* __restrict__ con,
    float* __restrict__ partials)
{
    __shared__ float sp[CHANS][HALO][HALO];   // sigmoid tile + halo, one branch
    __shared__ float swave[8];

    const int tid = threadIdx.x;
    const int tx  = tid & (TS - 1);
    const int ty  = tid >> 4;
    const int w0  = blockIdx.x * TS;
    const int h0  = blockIdx.y * TS;
    const int bb  = blockIdx.z;

    const int gw = w0 + tx;
    const int gh = h0 + ty;

    const size_t planeHW = (size_t)HH * WW;
    const size_t baseB   = (size_t)bb * CHANS * planeHW;
    const size_t pix     = (size_t)gh * WW + gw;

    const float invN8 = 1.0f / (float)((size_t)BATCH * CHANS * HH * WW);
    const float invN1 = 1.0f / (float)((size_t)BATCH * HH * WW);

    // ---- halo-ring ownership: threads 0..67 each own one perimeter cell ----
    int riy = 0, rix = 0;
    if (tid < HALO) {                 // top row
        riy = 0;           rix = tid;
    } else if (tid < 2 * HALO) {      // bottom row
        riy = HALO - 1;    rix = tid - HALO;
    } else {                          // left/right columns
        const int k = tid - 2 * HALO; // 0..31
        riy = 1 + (k >> 1);
        rix = (k & 1) ? (HALO - 1) : 0;
    }
    const int  rgy  = h0 - 1 + riy;
    const int  rgx  = w0 - 1 + rix;
    const bool ring = (tid < RING);
    const bool rin  = ring && ((unsigned)rgy < (unsigned)HH)
                           && ((unsigned)rgx < (unsigned)WW);
    const size_t roff = (size_t)rgy * WW + rgx;   // used only when rin

    // --- per-pixel invariants: connectivity target, edge mask, target -------
    float conv[CHANS];
    float sumc = 0.0f;
#pragma unroll
    for (int c = 0; c < CHANS; ++c) {
        conv[c] = con[baseB + (size_t)c * planeHW + pix];
        sumc += conv[c];
    }
    const float edge = (sumc > 0.0f && sumc < 8.0f) ? 1.0f : 0.0f;
    const float tval = target[(size_t)bb * planeHW + pix];

    // prefetch the dets tile into cache while the atts branch computes
#pragma unroll
    for (int c = 0; c < CHANS; ++c)
        __builtin_prefetch(dets + baseB + (size_t)c * planeHW + pix, 0, 3);

    float acc = 0.0f;   // weighted sum of all BCE element terms (negated later)

    for (int br = 0; br < 2; ++br) {
        const float* __restrict__ base = (br ? dets : atts) + baseB;

        __syncthreads();   // previous branch's LDS reads complete

        // ---- center 16x16: batch 8 channel loads, keep sigmoids in regs ----
        float pcv[CHANS];
        {
            float raw[CHANS];
#pragma unroll
            for (int c = 0; c < CHANS; ++c)
                raw[c] = base[(size_t)c * planeHW + pix];   // 8 loads in flight
#pragma unroll
            for (int c = 0; c < CHANS; ++c) {
                pcv[c] = fast_sigmoid(raw[c]);
                sp[c][ty + 1][tx + 1] = pcv[c];
            }
        }

        // ---- halo ring: one cell per thread, bounds check hoisted ----------
        if (ring) {
            float hp[CHANS];
#pragma unroll
            for (int c = 0; c < CHANS; ++c) hp[c] = 0.0f;   // zero padding
            if (rin) {
                float raw[CHANS];
#pragma unroll
                for (int c = 0; c < CHANS; ++c)
                    raw[c] = base[(size_t)c * planeHW + roff];  // batched
#pragma unroll
                for (int c = 0; c < CHANS; ++c) hp[c] = fast_sigmoid(raw[c]);
            }
#pragma unroll
            for (int c = 0; c < CHANS; ++c) sp[c][riy][rix] = hp[c];
        }
        __syncthreads();

        const int ly = ty + 1, lx = tx + 1;

        // vote_out stacking order: [a7, a3, a5, a1, a2, a6, a4, a8]
        float v[CHANS];
        v[0] = pcv[0] * sp[7][ly - 1][lx - 1];   // a7 = c0 * c7[r-1,w-1]
        v[1] = pcv[1] * sp[6][ly - 1][lx    ];   // a3 = c1 * c6[r-1,w]
        v[2] = pcv[2] * sp[5][ly - 1][lx + 1];   // a5 = c2 * c5[r-1,w+1]
        v[3] = pcv[3] * sp[4][ly    ][lx - 1];   // a1 = c3 * c4[r,w-1]
        v[4] = pcv[4] * sp[3][ly    ][lx + 1];   // a2 = c4 * c3[r,w+1]
        v[5] = pcv[5] * sp[2][ly + 1][lx - 1];   // a6 = c5 * c2[r+1,w-1]
        v[6] = pcv[6] * sp[1][ly + 1][lx    ];   // a4 = c6 * c1[r+1,w]
        v[7] = pcv[7] * sp[0][ly + 1][lx + 1];   // a8 = c7 * c0[r+1,w+1]

        float chsum = 0.0f, vsum = 0.0f, vmin = v[0];
#pragma unroll
        for (int c = 0; c < CHANS; ++c) {
            // 0.2*bicon + 0.8*conn, shared per-element scale 1/N8
            chsum = fmaf(0.2f, bce_term(v[c],   conv[c]), chsum);
            chsum = fmaf(0.8f, bce_term(pcv[c], conv[c]), chsum);
            vsum += v[c];
            vmin  = fminf(vmin, v[c]);
        }
        acc = fmaf(chsum, invN8, acc);

        const float glo = vsum * 0.125f;
        float pmap;
        if (br == 0) {
            pmap = glo;                                          // bce_loss1
        } else {
            // decouple_map = glo2*(1-edge) + (1-min(vote2))*edge
            pmap = fmaf(glo, 1.0f - edge, (1.0f - vmin) * edge); // de_loss2
        }
        acc = fmaf(bce_term(pmap, tval), invN1, acc);
    }

    // ---- wave32 reduction via WMMA, then 8-wave block reduction ------------
    const float wsum = wave_sum_wmma(acc);
    if ((tid & 31) == 0) swave[tid >> 5] = wsum;
    __syncthreads();
    if (tid == 0) {
        float tsum = 0.0f;
#pragma unroll
        for (int i = 0; i < 8; ++i) tsum += swave[i];
        const int blk = (bb * gridDim.y + blockIdx.y) * gridDim.x + blockIdx.x;
        partials[blk] = tsum;
    }
}

__global__ __launch_bounds__(256) void bicon_loss_final_reduce(
    const float* __restrict__ partials, float* __restrict__ out, int n)
{
    __shared__ float swave[8];
    const int tid = threadIdx.x;
    float acc = 0.0f;
    for (int i = tid; i < n; i += 256) acc += partials[i];  // fixed order
    const float wsum = wave_sum_wmma(acc);
    if ((tid & 31) == 0) swave[tid >> 5] = wsum;
    __syncthreads();
    if (tid == 0) {
        float tsum = 0.0f;
#pragma unroll
        for (int i = 0; i < 8; ++i) tsum += swave[i];
        out[0] = -tsum;   // loss = -(weighted sum of BCE terms)
    }
}

extern "C" void kernel_launch(void* const* d_in, const int* in_sizes, int n_in,
                              void* d_out, int out_size, void* d_ws, size_t ws_size,
                              hipStream_t stream) {
    const float* atts   = (const float*)d_in[0];
    const float* dets   = (const float*)d_in[1];
    const float* target = (const float*)d_in[2];
    const float* con    = (const float*)d_in[3];
    float* out      = (float*)d_out;
    float* partials = (float*)d_ws;   // NBLOCKS floats = 32 KiB

    dim3 grid(WW / TS, HH / TS, BATCH);   // (32, 32, 8)
    bicon_loss_fused_kernel<<<grid, 256, 0, stream>>>(atts, dets, target, con, partials);
    bicon_loss_final_reduce<<<1, 256, 0, stream>>>(partials, out, NBLOCKS);
}